// MultiHeadGraphAttention_34815004902138
// MI455X (gfx1250) — compile-verified
//
#include <hip/hip_runtime.h>
#include <hip/hip_bf16.h>

#define NB 4        // batch
#define NN 10000    // nodes
#define NE 170000   // edges
#define FIN 256
#define FF 128
#define NH 4
#define SLOPE 0.2f

#define KT 32               // K-chunk staged in LDS
#define LDS_PITCH 520       // 512 cols + 8 pad floats (8 banks) -> conflict-free halves

typedef float v2f __attribute__((ext_vector_type(2)));
typedef float v8f __attribute__((ext_vector_type(8)));

// D = A(16x4 f32) * B(4x16 f32) + C(16x16 f32), full f32 precision
#define WMMA_F32X4(a, b, c) \
    __builtin_amdgcn_wmma_f32_16x16x4_f32(false, (a), false, (b), (short)0, (c), false, false)

// ---- gfx1250 async global->LDS copy (portable inline asm; ASYNCcnt tracked) ----
__device__ __forceinline__ void async_load_b128(uint32_t lds_byte_off, const void* gaddr)
{
    uint64_t ga = (uint64_t)(uintptr_t)gaddr;
    asm volatile("global_load_async_to_lds_b128 %0, %1, off"
                 :: "v"(lds_byte_off), "v"(ga) : "memory");
}
__device__ __forceinline__ void wait_asynccnt0()
{
#if __has_builtin(__builtin_amdgcn_s_wait_asynccnt)
    __builtin_amdgcn_s_wait_asynccnt(0);
#else
    asm volatile("s_wait_asynccnt 0x0" ::: "memory");
#endif
}

// ---------------------------------------------------------------------------
// Kernel 1: h[hi][n][b][f] = sum_k x[b][n][k] * W_mlp[hi][k][f] + b_mlp[hi][f]
// M = B*N = 40000 rows, K = 256, Ncols = H*F = 512 (heads fused).
// Block = 8 waves, computes one 16-row M tile across all 512 columns.
// W K-chunks (32 x 512) are staged in LDS once per block with async b128
// copies; each wave then feeds WMMAs from ds_load (bank-conflict-free).
// ---------------------------------------------------------------------------
__global__ __launch_bounds__(256) void gemm_h_kernel(
    const float* __restrict__ x,      // (B, N, FIN)
    const float* __restrict__ W,      // (H, FIN, FF)
    const float* __restrict__ bias,   // (H, FF)
    float* __restrict__ h_ws)         // (H, N, B, FF)
{
    __shared__ float lds_w[KT * LDS_PITCH];   // 66,560 B

    const int t    = threadIdx.x;
    const int lane = t & 31;
    const int wave = t >> 5;
    const int m0   = blockIdx.x * 16;      // 2500 blocks
    const int n0   = wave * 64;            // column strip base (0..448)
    const int row  = lane & 15;
    const int kh   = (lane >> 4) * 2;      // 0 for lanes 0-15, 2 for lanes 16-31

    v8f acc[4];
    #pragma unroll
    for (int j = 0; j < 4; ++j) acc[j] = (v8f){0,0,0,0,0,0,0,0};

    int colj[4];
    #pragma unroll
    for (int j = 0; j < 4; ++j) colj[j] = n0 + j * 16 + row;   // fused col 0..511

    const size_t xbase = (size_t)(m0 + row) * FIN + kh;

    for (int kc = 0; kc < FIN / KT; ++kc) {
        const int k0 = kc * KT;

        // ---- stage W[k0..k0+31][0..511] into LDS (async, b128 per lane) ----
        // float4 element q: krow = q>>7, col = (q&127)*4 (never crosses a head)
        #pragma unroll
        for (int rnd = 0; rnd < (KT * 512 / 4) / 256; ++rnd) {   // 16 rounds
            const int q    = rnd * 256 + t;
            const int krow = q >> 7;
            const int col  = (q & 127) * 4;
            const int hi   = col >> 7;
            const int f    = col & 127;
            const float* gp = W + (size_t)hi * (FIN * FF) + (size_t)(k0 + krow) * FF + f;
            const uint32_t loff =
                (uint32_t)(uintptr_t)&lds_w[krow * LDS_PITCH + col];
            async_load_b128(loff, gp);
        }
        wait_asynccnt0();
        __syncthreads();

        // ---- consume: 8 WMMA K-steps of 4 ----
        #pragma unroll 2
        for (int ks = 0; ks < KT / 4; ++ks) {
            const int k    = k0 + ks * 4;
            const int krow = ks * 4 + kh;          // local K row pair base
            v2f a = *(const v2f*)(x + xbase + k);  // A frag (aligned float2)
            #pragma unroll
            for (int j = 0; j < 4; ++j) {
                v2f b;
                b.x = lds_w[(krow    ) * LDS_PITCH + colj[j]];
                b.y = lds_w[(krow + 1) * LDS_PITCH + colj[j]];
                acc[j] = WMMA_F32X4(a, b, acc[j]);
            }
        }
        __syncthreads();   // protect LDS before next chunk overwrites
    }

    // Store: C layout -> VGPR r: lanes0-15 row r, lanes16-31 row r+8
    const int mhalf = (lane >> 4) * 8;
    #pragma unroll
    for (int j = 0; j < 4; ++j) {
        const int col = colj[j];
        const int hi  = col >> 7;
        const int f   = col & 127;
        const float bv = bias[hi * FF + f];
        #pragma unroll
        for (int r = 0; r < 8; ++r) {
            const int m  = m0 + r + mhalf;       // row in (B*N)
            const int bb = m / NN;
            const int n  = m - bb * NN;
            h_ws[(((size_t)hi * NN + n) * NB + bb) * FF + f] = acc[j][r] + bv;
        }
    }
}

// ---------------------------------------------------------------------------
// Kernel 2: p_src[hi][n] = <h[hi][n][B-1][:], a_src[hi]>, same for p_dst.
// One wave per (hi, n): 32 lanes x float4, shfl reduction.
// ---------------------------------------------------------------------------
__global__ __launch_bounds__(256) void proj_kernel(
    const float* __restrict__ h_ws,    // (H, N, B, FF)
    const float* __restrict__ W_attn,  // (H, 2*FF, 1)
    float* __restrict__ p_src,
    float* __restrict__ p_dst)
{
    const int gw = blockIdx.x * 8 + (threadIdx.x >> 5);   // 0 .. H*N-1 (5000 blocks)
    const int lane = threadIdx.x & 31;
    const int hi = gw / NN;
    const int n  = gw - hi * NN;

    const float4 h4 = *((const float4*)(h_ws + (((size_t)hi * NN + n) * NB + (NB - 1)) * FF) + lane);
    const float4 a4 = *((const float4*)(W_attn + hi * 2 * FF) + lane);
    const float4 d4 = *((const float4*)(W_attn + hi * 2 * FF + FF) + lane);

    float ps = h4.x * a4.x + h4.y * a4.y + h4.z * a4.z + h4.w * a4.w;
    float pd = h4.x * d4.x + h4.y * d4.y + h4.z * d4.z + h4.w * d4.w;
    #pragma unroll
    for (int off = 16; off; off >>= 1) {
        ps += __shfl_xor(ps, off);
        pd += __shfl_xor(pd, off);
    }
    if (lane == 0) {
        p_src[gw] = ps;
        p_dst[gw] = pd;
    }
}

// ---------------------------------------------------------------------------
// Kernel 3: edge_start[n] = lower_bound(src, n)  (src is sorted ascending)
// ---------------------------------------------------------------------------
__global__ void edge_start_kernel(const int* __restrict__ src, int* __restrict__ edge_start)
{
    int n = blockIdx.x * 256 + threadIdx.x;
    if (n > NN) return;
    int lo = 0, hi = NE;
    while (lo < hi) {
        int mid = (lo + hi) >> 1;
        if (src[mid] < n) lo = mid + 1; else hi = mid;
    }
    edge_start[n] = lo;
}

__device__ __forceinline__ float edge_w(float s)
{
    s = (s > 0.f) ? s : SLOPE * s;                 // leaky relu
    s = fminf(fmaxf(s, -2.f), 2.f);                // clip
    return __expf(s);
}

// ---------------------------------------------------------------------------
// Kernel 4: per-node softmax-normalized aggregation (src sorted => no atomics).
// One block (256 thr) per node n. Thread t owns:
//   hi = (t>>5)>>1, b0 = ((t>>5)&1)*2 (two batches), f0 = (t&31)*4 (float4)
// => 8 output floats/thread * 256 = 2048 = B*H*F per node. Deterministic.
// ---------------------------------------------------------------------------
__global__ __launch_bounds__(256) void agg_kernel(
    const float* __restrict__ h_ws,    // (H, N, B, FF)
    const float* __restrict__ p_src,   // (H, N)
    const float* __restrict__ p_dst,   // (H, N)
    const int*   __restrict__ dst,
    const int*   __restrict__ edge_start,
    float* __restrict__ out)           // (B, N, H*FF)
{
    __shared__ float wave_part[8][NH];
    __shared__ float denom_sh[NH];

    const int n  = blockIdx.x;
    const int t  = threadIdx.x;
    const int lane = t & 31;
    const int wave = t >> 5;
    const int s0 = edge_start[n];
    const int s1 = edge_start[n + 1];

    // ---- phase A: denominators per head (deterministic reduction) ----
    float psv[NH];
    #pragma unroll
    for (int hh = 0; hh < NH; ++hh) psv[hh] = p_src[hh * NN + n];

    float part[NH] = {0.f, 0.f, 0.f, 0.f};
    for (int e = s0 + t; e < s1; e += 256) {
        const int d = dst[e];
        #pragma unroll
        for (int hh = 0; hh < NH; ++hh)
            part[hh] += edge_w(psv[hh] + p_dst[hh * NN + d]);
    }
    #pragma unroll
    for (int hh = 0; hh < NH; ++hh) {
        #pragma unroll
        for (int off = 16; off; off >>= 1) part[hh] += __shfl_xor(part[hh], off);
    }
    if (lane == 0) {
        #pragma unroll
        for (int hh = 0; hh < NH; ++hh) wave_part[wave][hh] = part[hh];
    }
    __syncthreads();
    if (t < NH) {
        float s = 0.f;
        #pragma unroll
        for (int w = 0; w < 8; ++w) s += wave_part[w][t];   // fixed order
        denom_sh[t] = s;
    }
    __syncthreads();

    // ---- phase B: weighted gather-accumulate in registers ----
    const int c   = t >> 5;
    const int hi  = c >> 1;
    const int b0  = (c & 1) * 2;
    const int f0  = (t & 31) * 4;

    const float ps   = psv[hi];
    const float dnm  = denom_sh[hi];
    const float inv  = (dnm > 0.f) ? (1.f / dnm) : 0.f;

    float4 acc0 = {0.f, 0.f, 0.f, 0.f};
    float4 acc1 = {0.f, 0.f, 0.f, 0.f};

    if (s1 > s0) {
        int d = dst[s0];
        for (int e = s0; e < s1; ++e) {
            const int dn = (e + 1 < s1) ? dst[e + 1] : d;
            // prefetch next edge's 2KB gather block (global_prefetch_b8)
            __builtin_prefetch(h_ws + (((size_t)hi * NN + dn) * NB + b0) * FF + f0, 0, 0);

            const float w = edge_w(ps + p_dst[hi * NN + d]) * inv;
            const size_t hb = (((size_t)hi * NN + d) * NB + b0) * FF + f0;
            const float4 v0 = *(const float4*)(h_ws + hb);
            const float4 v1 = *(const float4*)(h_ws + hb + FF);
            acc0.x = fmaf(v0.x, w, acc0.x); acc0.y = fmaf(v0.y, w, acc0.y);
            acc0.z = fmaf(v0.z, w, acc0.z); acc0.w = fmaf(v0.w, w, acc0.w);
            acc1.x = fmaf(v1.x, w, acc1.x); acc1.y = fmaf(v1.y, w, acc1.y);
            acc1.z = fmaf(v1.z, w, acc1.z); acc1.w = fmaf(v1.w, w, acc1.w);
            d = dn;
        }
    }

    const size_t obase = (size_t)hi * FF + f0;
    *(float4*)(out + ((size_t)(b0    ) * NN + n) * (NH * FF) + obase) = acc0;
    *(float4*)(out + ((size_t)(b0 + 1) * NN + n) * (NH * FF) + obase) = acc1;
}

// ---------------------------------------------------------------------------
extern "C" void kernel_launch(void* const* d_in, const int* in_sizes, int n_in,
                              void* d_out, int out_size, void* d_ws, size_t ws_size,
                              hipStream_t stream)
{
    const float* x      = (const float*)d_in[0];   // (B, N, FIN)
    const float* W_mlp  = (const float*)d_in[1];   // (H, FIN, FF)
    const float* b_mlp  = (const float*)d_in[2];   // (H, FF)
    const float* W_attn = (const float*)d_in[3];   // (H, 2FF, 1)
    const int*   src    = (const int*)d_in[4];     // (E,) sorted
    const int*   dst    = (const int*)d_in[5];     // (E,)
    float* out = (float*)d_out;

    float* ws      = (float*)d_ws;
    float* h_ws    = ws;                                   // H*N*B*FF = 20,480,000 f
    float* p_src   = ws + (size_t)NH * NN * NB * FF;       // 40,000 f
    float* p_dst   = p_src + (size_t)NH * NN;              // 40,000 f
    int*   e_start = (int*)(p_dst + (size_t)NH * NN);      // N+1 ints

    // 1) WMMA GEMM: h = x @ W_mlp + b  (heads fused into 512 columns)
    gemm_h_kernel<<<(NB * NN) / 16, 256, 0, stream>>>(x, W_mlp, b_mlp, h_ws);

    // 2) attention projections from last batch slice
    proj_kernel<<<(NH * NN) / 8, 256, 0, stream>>>(h_ws, W_attn, p_src, p_dst);

    // 3) per-node edge ranges (src is sorted)
    edge_start_kernel<<<(NN + 1 + 255) / 256, 256, 0, stream>>>(src, e_start);

    // 4) per-node softmax + aggregation, no atomics
    agg_kernel<<<NN, 256, 0, stream>>>(h_ws, p_src, p_dst, dst, e_start, out);
}